// GraphSAGEEncoder_31688268710106
// MI455X (gfx1250) — compile-verified
//
#include <hip/hip_runtime.h>
#include <hip/hip_bf16.h>

// ---------------------------------------------------------------------------
// GraphSAGE encoder for MI455X (gfx1250, wave32).
// GEMMs use V_WMMA_F32_16X16X4_F32 (fp32 matrix core path); aggregation is
// atomic-add scatter (memory-bound part, ~0.6 GB of edge traffic dominates).
// ---------------------------------------------------------------------------

typedef float v2f __attribute__((ext_vector_type(2)));
typedef float v8f __attribute__((ext_vector_type(8)));

#define ACT_ELU  0
#define ACT_RELU 1
#define ACT_NONE 2

// ---------------- scratch init ----------------
__global__ void fill_f32(float* __restrict__ p, long long n, float v) {
    long long i = (long long)blockIdx.x * blockDim.x + threadIdx.x;
    long long stride = (long long)gridDim.x * blockDim.x;
    for (; i < n; i += stride) p[i] = v;
}

__global__ void fill_u32(unsigned* __restrict__ p, long long n, unsigned v) {
    long long i = (long long)blockIdx.x * blockDim.x + threadIdx.x;
    long long stride = (long long)gridDim.x * blockDim.x;
    for (; i < n; i += stride) p[i] = v;
}

// ---------------- degree count ----------------
__global__ void count_deg(const int* __restrict__ dst, float* __restrict__ cnt, int E) {
    int e = blockIdx.x * blockDim.x + threadIdx.x;
    if (e < E) atomicAdd(&cnt[dst[e]], 1.0f);
}

// ---------------- edge scatter-add: agg[dst] += x[src] ----------------
__global__ void scatter_add(const float* __restrict__ x, const int* __restrict__ src,
                            const int* __restrict__ dst, float* __restrict__ agg,
                            int E, int logC) {
    int tid = blockIdx.x * blockDim.x + threadIdx.x;
    int C = 1 << logC;
    if (tid >= (E << logC)) return;
    int e = tid >> logC;
    int c = tid & (C - 1);
    atomicAdd(&agg[((long long)dst[e] << logC) + c], x[((long long)src[e] << logC) + c]);
}

// ---------------- mean: agg[i][c] /= max(cnt[i],1) ----------------
__global__ void scale_rows(float* __restrict__ agg, const float* __restrict__ cnt,
                           long long total, int logC) {
    long long tid = (long long)blockIdx.x * blockDim.x + threadIdx.x;
    if (tid >= total) return;
    long long i = tid >> logC;
    agg[tid] *= 1.0f / fmaxf(cnt[i], 1.0f);
}

// ---------------------------------------------------------------------------
// Fused GEMM: out = act(A1 @ B1 [+ A2 @ B2] + bias)
// One wave computes a 16 x (NT*16) output strip using fp32 WMMA 16x16x4.
// A fragments (16x4 tile): lane l holds rows m=l&15, k-pair base kb=(l>>4)*2.
// B fragments (4x16 tile): lane l holds col n=l&15, same k-pair base.
// C/D 16x16: VGPR v, lane l -> row = v + 8*(l>>4), col = l&15.
// M must be a multiple of 16 (true for all calls here).
// ---------------------------------------------------------------------------
template <int NT>
__global__ __launch_bounds__(256)
void gemm_fused(const float* __restrict__ A1, const float* __restrict__ A2,
                const float* __restrict__ B1, const float* __restrict__ B2,
                const float* __restrict__ bias, float* __restrict__ out,
                int M, int K, int act) {
    const int Nout = NT * 16;
    int wave = (blockIdx.x * blockDim.x + threadIdx.x) >> 5;
    int lane = threadIdx.x & 31;
    int mtiles = M >> 4;
    if (wave >= mtiles) return;           // uniform per-wave branch: EXEC stays all-1s

    const int m0   = wave << 4;
    const int mrow = lane & 15;           // A row within tile / B,C column within tile
    const int kb   = (lane >> 4) << 1;    // k-pair base for this half-wave

    v8f acc[NT];
    const v8f vzero = {0.f, 0.f, 0.f, 0.f, 0.f, 0.f, 0.f, 0.f};
#pragma unroll
    for (int t = 0; t < NT; ++t) acc[t] = vzero;

    const long long arow = (long long)(m0 + mrow) * K;

    for (int k = 0; k < K; k += 4) {
        // ---- first operand pair: A1 @ B1 ----
        v2f a;
        a.x = A1[arow + k + kb];
        a.y = A1[arow + k + kb + 1];
#pragma unroll
        for (int t = 0; t < NT; ++t) {
            v2f b;
            b.x = B1[(long long)(k + kb) * Nout + t * 16 + mrow];
            b.y = B1[(long long)(k + kb + 1) * Nout + t * 16 + mrow];
            acc[t] = __builtin_amdgcn_wmma_f32_16x16x4_f32(
                false, a, false, b, (short)0, acc[t], false, false);
        }
        // ---- optional second operand pair: A2 @ B2 ----
        if (A2 != nullptr) {              // uniform branch
            v2f a2;
            a2.x = A2[arow + k + kb];
            a2.y = A2[arow + k + kb + 1];
#pragma unroll
            for (int t = 0; t < NT; ++t) {
                v2f b2;
                b2.x = B2[(long long)(k + kb) * Nout + t * 16 + mrow];
                b2.y = B2[(long long)(k + kb + 1) * Nout + t * 16 + mrow];
                acc[t] = __builtin_amdgcn_wmma_f32_16x16x4_f32(
                    false, a2, false, b2, (short)0, acc[t], false, false);
            }
        }
    }

    // ---- epilogue: bias + activation + store ----
    const int rbase = m0 + ((lane >> 4) << 3);   // +8 for lanes 16..31
#pragma unroll
    for (int t = 0; t < NT; ++t) {
        int col = t * 16 + mrow;
        float bv = bias ? bias[col] : 0.0f;
#pragma unroll
        for (int v = 0; v < 8; ++v) {
            float val = acc[t][v] + bv;
            if (act == ACT_ELU)       val = val > 0.0f ? val : (__expf(val) - 1.0f);
            else if (act == ACT_RELU) val = fmaxf(val, 0.0f);
            out[(long long)(rbase + v) * Nout + col] = val;
        }
    }
}

// ---------------- per-graph mean+max pooling (H = 128) ----------------
__device__ __forceinline__ unsigned enc_f32(float f) {
    unsigned b = __float_as_uint(f);
    return (b & 0x80000000u) ? ~b : (b | 0x80000000u);   // monotone map
}
__device__ __forceinline__ float dec_f32(unsigned k) {
    unsigned b = (k & 0x80000000u) ? (k ^ 0x80000000u) : ~k;
    return __uint_as_float(b);
}

__global__ void pool_acc(const float* __restrict__ h, const int* __restrict__ batch,
                         float* __restrict__ gsum, unsigned* __restrict__ gmaxu,
                         float* __restrict__ gcnt, int N) {
    int tid = blockIdx.x * blockDim.x + threadIdx.x;
    if (tid >= (N << 7)) return;
    int i = tid >> 7;
    int c = tid & 127;
    int g = batch[i];
    float v = h[tid];
    atomicAdd(&gsum[(g << 7) + c], v);
    atomicMax(&gmaxu[(g << 7) + c], enc_f32(v));
    if (c == 0) atomicAdd(&gcnt[g], 1.0f);
}

__global__ void pool_fin(const float* __restrict__ gsum, const unsigned* __restrict__ gmaxu,
                         const float* __restrict__ gcnt, float* __restrict__ pool, int total) {
    int tid = blockIdx.x * blockDim.x + threadIdx.x;
    if (tid >= total) return;
    int g = tid >> 7;
    pool[tid] = gsum[tid] / fmaxf(gcnt[g], 1.0f) + dec_f32(gmaxu[tid]);
}

// ---------------------------------------------------------------------------
extern "C" void kernel_launch(void* const* d_in, const int* in_sizes, int n_in,
                              void* d_out, int out_size, void* d_ws, size_t ws_size,
                              hipStream_t stream) {
    const float* x   = (const float*)d_in[0];
    const int*   ei  = (const int*)d_in[1];
    const int*   bat = (const int*)d_in[2];
    const float* W1l = (const float*)d_in[3];
    const float* W1r = (const float*)d_in[4];
    const float* b1  = (const float*)d_in[5];
    const float* W2l = (const float*)d_in[6];
    const float* W2r = (const float*)d_in[7];
    const float* b2  = (const float*)d_in[8];
    const float* W3  = (const float*)d_in[9];
    const float* b3  = (const float*)d_in[10];
    const float* W4  = (const float*)d_in[11];
    const float* b4  = (const float*)d_in[12];
    float* out = (float*)d_out;

    const int IN = 64, H = 128, G = 64;
    const int N = in_sizes[0] / IN;       // 50000
    const int E = in_sizes[1] / 2;        // 800000
    const int* src = ei;
    const int* dst = ei + E;

    // ---- workspace layout (floats) ----
    float* ws = (float*)d_ws;
    long long off = 0;
    float*    cnt   = ws + off; off += N;                    // zeroed
    float*    agg1  = ws + off; off += (long long)N * IN;    // zeroed
    float*    agg2  = ws + off; off += (long long)N * H;     // zeroed
    float*    gsum  = ws + off; off += (long long)G * H;     // zeroed
    float*    gcnt  = ws + off; off += G;                    // zeroed
    const long long ZTOT = off;
    unsigned* gmaxu = (unsigned*)(ws + off); off += (long long)G * H;  // -> enc(-inf)
    float*    h1    = ws + off; off += (long long)N * H;
    float*    h2    = ws + off; off += (long long)N * H;
    float*    pool  = ws + off; off += (long long)G * H;
    float*    tmid  = ws + off; off += (long long)G * H;

    // 1) init scratch
    fill_f32<<<2048, 256, 0, stream>>>(ws, ZTOT, 0.0f);
    fill_u32<<<64, 256, 0, stream>>>(gmaxu, (long long)G * H, 0x007FFFFFu); // enc(-inf)

    // 2) degree counts
    count_deg<<<(E + 255) / 256, 256, 0, stream>>>(dst, cnt, E);

    // 3) layer 1: scatter mean + fused GEMM + ELU
    scatter_add<<<((E * IN) + 255) / 256, 256, 0, stream>>>(x, src, dst, agg1, E, 6);
    scale_rows<<<(int)(((long long)N * IN + 255) / 256), 256, 0, stream>>>(
        agg1, cnt, (long long)N * IN, 6);
    {   // h1 = elu(agg1 @ W1l + x @ W1r + b1), M=N, K=64, Nout=128
        int waves = N / 16, threads = waves * 32;
        gemm_fused<8><<<(threads + 255) / 256, 256, 0, stream>>>(
            agg1, x, W1l, W1r, b1, h1, N, IN, ACT_ELU);
    }

    // 4) layer 2
    scatter_add<<<(int)(((long long)E * H + 255) / 256), 256, 0, stream>>>(
        h1, src, dst, agg2, E, 7);
    scale_rows<<<(int)(((long long)N * H + 255) / 256), 256, 0, stream>>>(
        agg2, cnt, (long long)N * H, 7);
    {   // h2 = elu(agg2 @ W2l + h1 @ W2r + b2), M=N, K=128, Nout=128
        int waves = N / 16, threads = waves * 32;
        gemm_fused<8><<<(threads + 255) / 256, 256, 0, stream>>>(
            agg2, h1, W2l, W2r, b2, h2, N, H, ACT_ELU);
    }

    // 5) per-graph mean+max pool
    pool_acc<<<(int)(((long long)N * H + 255) / 256), 256, 0, stream>>>(
        h2, bat, gsum, gmaxu, gcnt, N);
    pool_fin<<<(G * H + 255) / 256, 256, 0, stream>>>(gsum, gmaxu, gcnt, pool, G * H);

    // 6) MLP head: tmid = relu(pool @ W3 + b3); out = tmid @ W4 + b4
    {
        int waves = G / 16, threads = waves * 32;   // 4 waves
        gemm_fused<8><<<1, ((threads + 255) / 256) * 256 ? 256 : 256, 0, stream>>>(
            pool, nullptr, W3, nullptr, b3, tmid, G, H, ACT_RELU);
        gemm_fused<4><<<1, 256, 0, stream>>>(
            tmid, nullptr, W4, nullptr, b4, out, G, H, ACT_NONE);
        (void)threads;
    }
}